// LearnableGate16Array_1494648619383
// MI455X (gfx1250) — compile-verified
//
#include <hip/hip_runtime.h>

// CDNA5 / gfx1250. Wave32.
typedef __attribute__((ext_vector_type(2))) float v2f;
typedef __attribute__((ext_vector_type(4))) float v4f;
typedef __attribute__((ext_vector_type(8))) float v8f;

// W16_TO_4 from the reference (row 3 per NumPy assignment order).
constexpr float kW16[4][16] = {
  { 0, 0, 0, 0, 0, 0, 0, 0,  1, 1, 1, 1, 1, 1, 1, 1},
  { 0, 0, 1, 1, 0, 0, 1, 1, -1,-1, 0, 0,-1,-1, 0, 0},
  { 0, 0, 0, 0, 1, 1, 1, 1, -1,-1,-1,-1, 0, 0, 0, 0},
  { 0, 1,-1, 0,-1, 0,-2,-1,  1, 2, 0, 1, 0, 1,-1, 0},
};

// Compile-time A-matrix lane layout for V_WMMA_F32_16X16X4_F32 (ISA 7.12.2):
// A is 16x4 f32 in 2 VGPRs: lanes 0-15 hold K={0,1}, lanes 16-31 hold K={2,3},
// with M = lane%16. Rows 4..15 of the padded W matrix are zero.
// cAtab[i][lane] = {A[m][4i+kh], A[m][4i+kh+1]}.
struct ATab { float v[4][32][2]; };
constexpr ATab make_atab() {
  ATab t{};
  for (int i = 0; i < 4; ++i)
    for (int lane = 0; lane < 32; ++lane) {
      const int m  = lane & 15;
      const int kh = (lane < 16) ? 0 : 2;
      t.v[i][lane][0] = (m < 4) ? kW16[m][4 * i + kh]     : 0.0f;
      t.v[i][lane][1] = (m < 4) ? kW16[m][4 * i + kh + 1] : 0.0f;
    }
  return t;
}
__constant__ ATab cAtab = make_atab();

// Stage 1: wc4[n] = {wc0..wc3}[n] = W16_TO_4 @ softmax(w[:,n]).
// One wave32 per 16-column tile, 4 chained V_WMMA_F32_16X16X4_F32 (K=16).
__global__ void wc_wmma_kernel(const float* __restrict__ w,
                               v4f* __restrict__ wc4, int N) {
  const int lane = threadIdx.x & 31;
  const int wave = (int)((blockIdx.x * (unsigned)blockDim.x + threadIdx.x) >> 5);
  const int n0 = wave << 4;               // wave-uniform tile start
  if (n0 >= N) return;                    // wave-uniform exit; EXEC all-1s inside
  const int mrow = lane & 15;
  const int col  = n0 + mrow;
  const bool lo  = (lane < 16);

  // Every lane (both EXEC halves) softmaxes column n0+(lane%16) so the
  // B-matrix rows each half must supply are locally available (no shuffles).
  float wv[16];
  float mx = -3.402823466e38f;
#pragma unroll
  for (int r = 0; r < 16; ++r) { wv[r] = w[(size_t)r * N + col]; mx = fmaxf(mx, wv[r]); }
  float s = 0.f;
#pragma unroll
  for (int r = 0; r < 16; ++r) { wv[r] = __expf(wv[r] - mx); s += wv[r]; }
  const float inv = 1.0f / s;
#pragma unroll
  for (int r = 0; r < 16; ++r) wv[r] *= inv;

  v8f acc = {};
#pragma unroll
  for (int i = 0; i < 4; ++i) {
    // A chunk: one coalesced 8B constant-memory load per lane (precomputed layout).
    v2f a = *(const v2f*)cAtab.v[i][lane];
    // B chunk: 4x16 tile, lanes 0-15 carry K={4i,4i+1}, lanes 16-31 K={4i+2,4i+3}.
    // Constant register indices + one v_cndmask per component.
    v2f b;
    b.x = lo ? wv[4 * i + 0] : wv[4 * i + 2];
    b.y = lo ? wv[4 * i + 1] : wv[4 * i + 3];
    acc = __builtin_amdgcn_wmma_f32_16x16x4_f32(
        /*neg_a=*/false, a, /*neg_b=*/false, b,
        /*c_mod=*/(short)0, acc, /*reuse_a=*/false, /*reuse_b=*/false);
  }

  // D layout: VGPR r = row M=r for lanes 0-15; rows 0..3 are wc0..wc3.
  if (lo) {
    v4f o;
    o.x = acc[0]; o.y = acc[1]; o.z = acc[2]; o.w = acc[3];
    wc4[col] = o;
  }
}

__device__ __forceinline__ v4f gate4(v4f x0, v4f x1, v4f c0, v4f c1, v4f c2, v4f c3) {
  v4f o;
  o.x = fmaf(c0.w, x0.x * x0.y, fmaf(c0.z, x0.y, fmaf(c0.y, x0.x, c0.x)));
  o.y = fmaf(c1.w, x0.z * x0.w, fmaf(c1.z, x0.w, fmaf(c1.y, x0.z, c1.x)));
  o.z = fmaf(c2.w, x1.x * x1.y, fmaf(c2.z, x1.y, fmaf(c2.y, x1.x, c2.x)));
  o.w = fmaf(c3.w, x1.z * x1.w, fmaf(c3.z, x1.w, fmaf(c3.y, x1.z, c3.x)));
  return o;
}

// Stage 2: pure HBM streaming. One thread = 4 columns x 2 batch rows
// (reuses the wc4 loads across both rows -> halves L2 coefficient traffic).
// x row layout: (A,B) pairs interleaved, so 8 consecutive floats = 4 pairs.
__global__ void gate_stream_kernel(const float* __restrict__ x,
                                   const v4f* __restrict__ wc4,
                                   float* __restrict__ out, int N, int batch) {
  const int b0 = blockIdx.y << 1;
  const int ng = blockIdx.x * blockDim.x + threadIdx.x;
  const int n  = ng << 2;
  if (n >= N) return;

  v4f c0 = wc4[n + 0];   // cached: wc stays hot in L2 across all batch rows
  v4f c1 = wc4[n + 1];
  v4f c2 = wc4[n + 2];
  v4f c3 = wc4[n + 3];

  const size_t xrow = 2u * (size_t)N;
  const v4f* xp0 = (const v4f*)(x + (size_t)b0 * xrow + 2u * (size_t)n);
  v4f x0 = __builtin_nontemporal_load(xp0);      // row b0: pairs n, n+1
  v4f x1 = __builtin_nontemporal_load(xp0 + 1);  // row b0: pairs n+2, n+3
  v4f o0 = gate4(x0, x1, c0, c1, c2, c3);
  __builtin_nontemporal_store(o0, (v4f*)(out + (size_t)b0 * (size_t)N + (size_t)n));

  if (b0 + 1 < batch) {
    const v4f* xp1 = (const v4f*)((const float*)xp0 + xrow);
    v4f y0 = __builtin_nontemporal_load(xp1);
    v4f y1 = __builtin_nontemporal_load(xp1 + 1);
    v4f o1 = gate4(y0, y1, c0, c1, c2, c3);
    __builtin_nontemporal_store(o1, (v4f*)(out + (size_t)(b0 + 1) * (size_t)N + (size_t)n));
  }
}

extern "C" void kernel_launch(void* const* d_in, const int* in_sizes, int n_in,
                              void* d_out, int out_size, void* d_ws, size_t ws_size,
                              hipStream_t stream) {
  const float* x = (const float*)d_in[0];   // (batch, 2N) f32
  const float* w = (const float*)d_in[1];   // (16, N) f32
  float* out = (float*)d_out;               // (batch, N) f32

  const int N = in_sizes[1] / 16;           // 65536
  const int batch = in_sizes[0] / (2 * N);  // 256
  v4f* wc4 = (v4f*)d_ws;                    // N * 16 bytes = 1 MiB scratch

  // Stage 1: N/16 waves, 8 waves per 256-thread block.
  {
    const int waves   = N / 16;
    const int threads = waves * 32;
    const int block   = 256;
    const int grid    = (threads + block - 1) / block;
    wc_wmma_kernel<<<grid, block, 0, stream>>>(w, wc4, N);
  }
  // Stage 2: memory-bound stream; grid.y covers batch in pairs of rows.
  {
    dim3 block(256, 1, 1);
    dim3 grid((unsigned)((N / 4 + 255) / 256), (unsigned)((batch + 1) / 2), 1);
    gate_stream_kernel<<<grid, block, 0, stream>>>(x, wc4, out, N, batch);
  }
}